// GCNN_3p_old_81063212744718
// MI455X (gfx1250) — compile-verified
//
#include <hip/hip_runtime.h>

// -------- problem constants (from the reference) --------
#define IN_F   128
#define C1     64
#define C2     64
#define OUTF   2
#define NGRAPH 128

typedef __attribute__((ext_vector_type(2))) float v2f;
typedef __attribute__((ext_vector_type(8))) float v8f;

// ---------------- utility kernels ----------------
__global__ void k_zero(float* __restrict__ p, int n) {
    int i = blockIdx.x * blockDim.x + threadIdx.x;
    int stride = gridDim.x * blockDim.x;
    for (; i < n; i += stride) p[i] = 0.0f;
}

// deg[dst[e]] += 1  (self loop added in k_dinv)
__global__ void k_degree(const int* __restrict__ dst, float* __restrict__ deg, int e) {
    int i = blockIdx.x * blockDim.x + threadIdx.x;
    if (i < e) unsafeAtomicAdd(&deg[dst[i]], 1.0f);
}

// dinv = rsqrt(deg + 1)   (+1 = self loop; always > 0)
__global__ void k_dinv(float* __restrict__ deg, int n) {
    int i = blockIdx.x * blockDim.x + threadIdx.x;
    if (i < n) deg[i] = __frsqrt_rn(deg[i] + 1.0f);
}

// ---------------- WMMA GEMM:  C[n x 64] = A[n x K] @ W[K x 64] ----------------
// 256 threads = 8 waves; each wave computes a 16x64 output tile using
// V_WMMA_F32_16X16X4_F32.  W is staged in LDS.
template <int K>
__global__ __launch_bounds__(256) void k_gemm_wmma(const float* __restrict__ A,
                                                   const float* __restrict__ W,
                                                   float* __restrict__ C, int n) {
    __shared__ float sW[K * 64];
    const int tid = threadIdx.x;
    for (int i = tid; i < K * 64; i += 256) sW[i] = W[i];
    __syncthreads();

    const int wave  = tid >> 5;
    const int lane  = tid & 31;
    const int row0  = blockIdx.x * 128 + wave * 16;
    if (row0 >= n) return;                 // uniform per wave

    const int lh  = lane >> 4;             // lane half: 0 | 1
    const int l16 = lane & 15;
    int rowA = row0 + l16;                 // A-frag row for this lane
    if (rowA >= n) rowA = n - 1;           // clamp (keeps EXEC all-1 for WMMA)

    v8f c0 = {}, c1 = {}, c2 = {}, c3 = {};

#pragma unroll
    for (int k = 0; k < K; k += 4) {
        const int kb = k + (lh << 1);      // this lane covers K = kb, kb+1
        // A fragment: 16x4 f32 -> 2 VGPRs per lane (float2, 8B aligned)
        const v2f a = *(const v2f*)(A + (long)rowA * K + kb);
        // B fragments for 4 column tiles, from LDS
        v2f b0, b1, b2, b3;
        b0.x = sW[kb * 64 +  0 + l16]; b0.y = sW[(kb + 1) * 64 +  0 + l16];
        b1.x = sW[kb * 64 + 16 + l16]; b1.y = sW[(kb + 1) * 64 + 16 + l16];
        b2.x = sW[kb * 64 + 32 + l16]; b2.y = sW[(kb + 1) * 64 + 32 + l16];
        b3.x = sW[kb * 64 + 48 + l16]; b3.y = sW[(kb + 1) * 64 + 48 + l16];

        c0 = __builtin_amdgcn_wmma_f32_16x16x4_f32(false, a, false, b0, (short)0, c0, false, false);
        c1 = __builtin_amdgcn_wmma_f32_16x16x4_f32(false, a, false, b1, (short)0, c1, false, false);
        c2 = __builtin_amdgcn_wmma_f32_16x16x4_f32(false, a, false, b2, (short)0, c2, false, false);
        c3 = __builtin_amdgcn_wmma_f32_16x16x4_f32(false, a, false, b3, (short)0, c3, false, false);
    }

    // C/D layout: VGPR v holds row M=v (lanes 0-15) / M=v+8 (lanes 16-31)
    const int rbase = row0 + (lh << 3);
#pragma unroll
    for (int v = 0; v < 8; ++v) {
        const int r = rbase + v;
        if (r < n) {
            float* p = C + (long)r * 64 + l16;
            p[0]  = c0[v];
            p[16] = c1[v];
            p[32] = c2[v];
            p[48] = c3[v];
        }
    }
}

// ---------------- edge scatter: acc[dst] += xw[src] * dinv[src]*dinv[dst] ----------------
// 16 threads per edge, float4 per thread.
__global__ void k_scatter(const int* __restrict__ src, const int* __restrict__ dst,
                          const float* __restrict__ dinv, const float* __restrict__ xw,
                          float* __restrict__ acc, int e) {
    int t = blockIdx.x * blockDim.x + threadIdx.x;
    int edge = t >> 4;
    if (edge >= e) return;
    int c4 = (t & 15) << 2;
    int s = src[edge], d = dst[edge];
    float nrm = dinv[s] * dinv[d];
    float4 v = *(const float4*)(xw + (long)s * 64 + c4);
    float* p = acc + (long)d * 64 + c4;
    unsafeAtomicAdd(p + 0, v.x * nrm);
    unsafeAtomicAdd(p + 1, v.y * nrm);
    unsafeAtomicAdd(p + 2, v.z * nrm);
    unsafeAtomicAdd(p + 3, v.w * nrm);
}

// ---------------- h = relu(acc + xw*dinv^2 + b) ----------------
__global__ void k_finalize(const float* __restrict__ acc, const float* xw,
                           const float* __restrict__ dinv, const float* __restrict__ b,
                           float* h, int n64) {
    int i = blockIdx.x * blockDim.x + threadIdx.x;
    if (i >= n64) return;
    int node = i >> 6, c = i & 63;
    float di = dinv[node];
    float v = acc[i] + xw[i] * di * di + b[c];
    h[i] = v > 0.0f ? v : 0.0f;
}

// ---------------- pooling atomics ----------------
__global__ void k_pool(const float* __restrict__ h, const int* __restrict__ batch,
                       float* __restrict__ psum, float* __restrict__ pmax, int n64) {
    int i = blockIdx.x * blockDim.x + threadIdx.x;
    if (i >= n64) return;
    int node = i >> 6, c = i & 63;
    int g = batch[node];
    float v = h[i];
    unsafeAtomicAdd(&psum[g * 64 + c], v);
    // h >= 0 after relu -> uint bit-pattern order == float order; init 0 matches ref
    atomicMax((unsigned int*)&pmax[g * 64 + c], __float_as_uint(v));
}

__global__ void k_count(const int* __restrict__ batch, float* __restrict__ cnt, int n) {
    int i = blockIdx.x * blockDim.x + threadIdx.x;
    if (i < n) unsafeAtomicAdd(&cnt[batch[i]], 1.0f);
}

// ---------------- per-graph head: xcat + xcat@Wlin + blin ----------------
__global__ __launch_bounds__(192) void k_head(const float* __restrict__ psum,
                                              const float* __restrict__ pmax,
                                              const float* __restrict__ pcnt,
                                              const float* __restrict__ Wlin,
                                              const float* __restrict__ blin,
                                              float* __restrict__ out) {
    __shared__ float xc[3 * C2];
    int g = blockIdx.x, t = threadIdx.x;
    float val;
    if (t < 64)        val = psum[g * 64 + t];
    else if (t < 128)  val = psum[g * 64 + (t - 64)] / fmaxf(pcnt[g], 1.0f);
    else               val = pmax[g * 64 + (t - 128)];
    xc[t] = val;
    out[g * (3 * C2) + t] = val;               // xcat, row-major [G, 192]
    __syncthreads();
    if (t < OUTF) {
        float a = blin[t];
#pragma unroll 8
        for (int k = 0; k < 3 * C2; ++k) a += xc[k] * Wlin[k * OUTF + t];
        out[NGRAPH * (3 * C2) + g * OUTF + t] = a;   // out, [G, 2]
    }
}

// ---------------- launcher ----------------
extern "C" void kernel_launch(void* const* d_in, const int* in_sizes, int n_in,
                              void* d_out, int out_size, void* d_ws, size_t ws_size,
                              hipStream_t stream) {
    const float* x    = (const float*)d_in[0];
    const float* W1   = (const float*)d_in[1];
    const float* b1   = (const float*)d_in[2];
    const float* W2   = (const float*)d_in[3];
    const float* b2   = (const float*)d_in[4];
    const float* Wlin = (const float*)d_in[5];
    const float* blin = (const float*)d_in[6];
    const int*   eidx = (const int*)d_in[7];
    const int*   batch= (const int*)d_in[8];

    const int N = in_sizes[0] / IN_F;     // 100000
    const int E = in_sizes[7] / 2;        // 1600000
    const int* src = eidx;
    const int* dst = eidx + E;

    // workspace layout (floats): dinv[N] | bufA[N*64] | bufB[N*64] | psum | pmax | pcnt
    float* ws   = (float*)d_ws;
    float* dinv = ws;
    float* bufA = dinv + N;
    float* bufB = bufA + (long)N * 64;
    float* psum = bufB + (long)N * 64;
    float* pmax = psum + NGRAPH * 64;
    float* pcnt = pmax + NGRAPH * 64;

    const int n64 = N * 64;
    const int TB = 256;
    const int gN    = (N + TB - 1) / TB;
    const int gE    = (E + TB - 1) / TB;
    const int gN64  = (n64 + TB - 1) / TB;
    const int gScat = (E * 16 + TB - 1) / TB;
    const int gGemm = (N + 127) / 128;
    const int gPool = (NGRAPH * 64 * 2 + NGRAPH + TB - 1) / TB;

    // 1. zero dinv, acc (bufB), pools
    k_zero<<<gN,   TB, 0, stream>>>(dinv, N);
    k_zero<<<gN64, TB, 0, stream>>>(bufB, n64);
    k_zero<<<gPool,TB, 0, stream>>>(psum, NGRAPH * 64 * 2 + NGRAPH);
    // 2. degrees + dinv
    k_degree<<<gE, TB, 0, stream>>>(dst, dinv, E);
    k_dinv<<<gN, TB, 0, stream>>>(dinv, N);
    // 3. layer 1
    k_gemm_wmma<IN_F><<<gGemm, TB, 0, stream>>>(x, W1, bufA, N);      // bufA = x@W1
    k_scatter<<<gScat, TB, 0, stream>>>(src, dst, dinv, bufA, bufB, E);
    k_finalize<<<gN64, TB, 0, stream>>>(bufB, bufA, dinv, b1, bufA, n64); // h1 -> bufA
    // 4. layer 2
    k_gemm_wmma<C1><<<gGemm, TB, 0, stream>>>(bufA, W2, bufB, N);     // bufB = h1@W2
    k_zero<<<gN64, TB, 0, stream>>>(bufA, n64);
    k_scatter<<<gScat, TB, 0, stream>>>(src, dst, dinv, bufB, bufA, E);
    k_finalize<<<gN64, TB, 0, stream>>>(bufA, bufB, dinv, b2, bufB, n64); // h2 -> bufB
    // 5. pooling + head
    k_count<<<gN, TB, 0, stream>>>(batch, pcnt, N);
    k_pool<<<gN64, TB, 0, stream>>>(bufB, batch, psum, pmax, n64);
    k_head<<<NGRAPH, 192, 0, stream>>>(psum, pmax, pcnt, Wlin, blin, (float*)d_out);
}